// MambaBlock_10788957847755
// MI455X (gfx1250) — compile-verified
//
#include <hip/hip_runtime.h>
#include <math.h>

#define IN_DIM   192
#define N_STATE  16
#define D_CONV   4
#define D_INNER  384
#define DT_RANK  12
#define BATCH    8
#define SEQ      4096
#define NTOK     (BATCH * SEQ)      /* 32768 */
#define XZ_DIM   (2 * D_INNER)      /* 768   */

typedef __attribute__((ext_vector_type(2))) float v2f;
typedef __attribute__((ext_vector_type(8))) float v8f;

__device__ __forceinline__ float silu_f(float x) { return x / (1.f + __expf(-x)); }

// ---------------------------------------------------------------------------
// 1) in-projection: xz = x @ W_in^T ; split columns into xs[:,0:384], res[:,384:768]
//    fp32 WMMA 16x16x4. One wave computes a 16x64 output strip (4 tiles, A reuse).
// ---------------------------------------------------------------------------
__global__ void k_gemm_in(const float* __restrict__ X, const float* __restrict__ W,
                          float* __restrict__ xs, float* __restrict__ res)
{
    const int NT = 4;
    const int K  = IN_DIM;
    const int ngroups = XZ_DIM / (16 * NT);           // 12
    int wave = (int)((blockIdx.x * blockDim.x + threadIdx.x) >> 5);
    int lane = threadIdx.x & 31;
    int mt = wave / ngroups;                          // wave-uniform
    int ng = wave % ngroups;
    if (mt >= NTOK / 16) return;                      // uniform -> EXEC stays all-1s
    int m0 = mt * 16, n0 = ng * 16 * NT;
    int row = m0 + (lane & 15);
    int kf  = (lane >> 4) * 2;                        // lanes 16-31 hold K+2,K+3

    v8f acc[NT] = {};
    for (int k0 = 0; k0 < K; k0 += 4) {
        v2f a = *(const v2f*)(X + (size_t)row * K + k0 + kf);
#pragma unroll
        for (int t = 0; t < NT; ++t) {
            int col = n0 + t * 16 + (lane & 15);
            v2f bb = *(const v2f*)(W + (size_t)col * K + k0 + kf);
            acc[t] = __builtin_amdgcn_wmma_f32_16x16x4_f32(
                false, a, false, bb, (short)0, acc[t], false, false);
        }
    }
    int rtop = m0 + (lane >> 4) * 8;                  // C/D: vgpr i -> rows m0+i / m0+8+i
#pragma unroll
    for (int t = 0; t < NT; ++t) {
        int c = n0 + t * 16 + (lane & 15);
        float* dst; int cc;
        if (c < D_INNER) { dst = xs;  cc = c; }
        else             { dst = res; cc = c - D_INNER; }
#pragma unroll
        for (int i = 0; i < 8; ++i)
            dst[(size_t)(rtop + i) * D_INNER + cc] = acc[t][i];
    }
}

// ---------------------------------------------------------------------------
// 2) depthwise causal conv (D_CONV=4) + bias + SiLU -> u
// ---------------------------------------------------------------------------
__global__ void k_conv_silu(const float* __restrict__ xs, const float* __restrict__ cw,
                            const float* __restrict__ cb, float* __restrict__ u)
{
    long idx = (long)blockIdx.x * blockDim.x + threadIdx.x;
    if (idx >= (long)NTOK * D_INNER) return;
    int  d   = (int)(idx % D_INNER);
    long tok = idx / D_INNER;
    int  t   = (int)(tok % SEQ);
    float acc = cb[d];
#pragma unroll
    for (int j = 0; j < D_CONV; ++j) {
        int tt = t - (D_CONV - 1) + j;
        if (tt >= 0)
            acc += xs[(tok - (D_CONV - 1) + j) * D_INNER + d] * cw[d * D_CONV + j];
    }
    u[idx] = silu_f(acc);
}

// ---------------------------------------------------------------------------
// 3) x_dbl = u @ W_x^T  (N=44) -> dt_low[12], B[16], C[16] per token
// ---------------------------------------------------------------------------
__global__ void k_xproj(const float* __restrict__ u, const float* __restrict__ Wx,
                        float* __restrict__ dtl, float* __restrict__ Bb,
                        float* __restrict__ Cb)
{
    long idx = (long)blockIdx.x * blockDim.x + threadIdx.x;
    const int NO = DT_RANK + 2 * N_STATE;             // 44
    if (idx >= (long)NTOK * NO) return;
    int  o   = (int)(idx % NO);
    long tok = idx / NO;
    const float* ur = u  + tok * D_INNER;
    const float* wr = Wx + (long)o * D_INNER;
    float acc = 0.f;
#pragma unroll 8
    for (int k = 0; k < D_INNER; ++k) acc = fmaf(ur[k], wr[k], acc);
    if (o < DT_RANK)                dtl[tok * DT_RANK + o] = acc;
    else if (o < DT_RANK + N_STATE) Bb[tok * N_STATE + (o - DT_RANK)] = acc;
    else                            Cb[tok * N_STATE + (o - DT_RANK - N_STATE)] = acc;
}

// ---------------------------------------------------------------------------
// 4) delta = softplus(dt_low @ W_dt^T + b_dt)
// ---------------------------------------------------------------------------
__global__ void k_dt(const float* __restrict__ dtl, const float* __restrict__ Wdt,
                     const float* __restrict__ bdt, float* __restrict__ delta)
{
    long idx = (long)blockIdx.x * blockDim.x + threadIdx.x;
    if (idx >= (long)NTOK * D_INNER) return;
    int  d   = (int)(idx % D_INNER);
    long tok = idx / D_INNER;
    float acc = bdt[d];
#pragma unroll
    for (int r = 0; r < DT_RANK; ++r)
        acc = fmaf(dtl[tok * DT_RANK + r], Wdt[d * DT_RANK + r], acc);
    delta[idx] = (acc > 20.f) ? acc : log1pf(__expf(acc));
}

// ---------------------------------------------------------------------------
// 5) selective scan: lane per (b,d,n); 16-lane shuffle reduction over n for y.
//    y = scan_y + u*D folded in here.
// ---------------------------------------------------------------------------
__global__ void k_scan(const float* __restrict__ delta, const float* __restrict__ u,
                       const float* __restrict__ Bb, const float* __restrict__ Cb,
                       const float* __restrict__ A_log, const float* __restrict__ Dv,
                       float* __restrict__ y)
{
    int gid = blockIdx.x * blockDim.x + threadIdx.x;   // [0, 8*384*16)
    int n  = gid & (N_STATE - 1);
    int ch = gid >> 4;                                  // 0..3071
    int b  = ch / D_INNER;
    int d  = ch % D_INNER;
    float A  = -__expf(A_log[d * N_STATE + n]);
    float Dd = Dv[d];
    float h = 0.f;
    long tokBase = (long)b * SEQ;
    for (int t = 0; t < SEQ; ++t) {
        long tok = tokBase + t;
        float dt = delta[tok * D_INNER + d];
        float ut = u[tok * D_INNER + d];
        float Bt = Bb[tok * N_STATE + n];
        float Ct = Cb[tok * N_STATE + n];
        h = h * __expf(dt * A) + (dt * ut) * Bt;
        float p = h * Ct;
        p += __shfl_xor(p, 1, 16);
        p += __shfl_xor(p, 2, 16);
        p += __shfl_xor(p, 4, 16);
        p += __shfl_xor(p, 8, 16);
        if (n == 0) y[tok * D_INNER + d] = p + ut * Dd;
    }
}

// ---------------------------------------------------------------------------
// 6) LayerNorm over d (population var) + SiLU(res) gate. Wave per token.
// ---------------------------------------------------------------------------
__global__ void k_ln(const float* __restrict__ y, const float* __restrict__ res,
                     const float* __restrict__ gamma, const float* __restrict__ beta,
                     float* __restrict__ yn)
{
    int wave = (int)((blockIdx.x * blockDim.x + threadIdx.x) >> 5);
    int lane = threadIdx.x & 31;
    if (wave >= NTOK) return;                          // wave-uniform
    const float* yr = y + (long)wave * D_INNER;
    float vals[D_INNER / 32];
    float s = 0.f, s2 = 0.f;
#pragma unroll
    for (int i = 0; i < D_INNER / 32; ++i) {
        float v = yr[lane + i * 32];
        vals[i] = v; s += v; s2 += v * v;
    }
#pragma unroll
    for (int m = 1; m < 32; m <<= 1) {
        s  += __shfl_xor(s,  m, 32);
        s2 += __shfl_xor(s2, m, 32);
    }
    float mu   = s  * (1.f / D_INNER);
    float var  = s2 * (1.f / D_INNER) - mu * mu;
    float rstd = rsqrtf(var + 1e-5f);
    const float* rr = res + (long)wave * D_INNER;
#pragma unroll
    for (int i = 0; i < D_INNER / 32; ++i) {
        int c = lane + i * 32;
        float g = (vals[i] - mu) * rstd * gamma[c] + beta[c];
        yn[(long)wave * D_INNER + c] = g * silu_f(rr[c]);
    }
}

// ---------------------------------------------------------------------------
// 7) out-projection: out = yn @ W_out^T (K=384, N=192), fp32 WMMA, 4 tiles/wave
// ---------------------------------------------------------------------------
__global__ void k_gemm_out(const float* __restrict__ Y, const float* __restrict__ W,
                           float* __restrict__ Out)
{
    const int NT = 4;
    const int K  = D_INNER;
    const int ngroups = IN_DIM / (16 * NT);            // 3
    int wave = (int)((blockIdx.x * blockDim.x + threadIdx.x) >> 5);
    int lane = threadIdx.x & 31;
    int mt = wave / ngroups;
    int ng = wave % ngroups;
    if (mt >= NTOK / 16) return;                       // uniform
    int m0 = mt * 16, n0 = ng * 16 * NT;
    int row = m0 + (lane & 15);
    int kf  = (lane >> 4) * 2;

    v8f acc[NT] = {};
    for (int k0 = 0; k0 < K; k0 += 4) {
        v2f a = *(const v2f*)(Y + (size_t)row * K + k0 + kf);
#pragma unroll
        for (int t = 0; t < NT; ++t) {
            int col = n0 + t * 16 + (lane & 15);
            v2f bb = *(const v2f*)(W + (size_t)col * K + k0 + kf);
            acc[t] = __builtin_amdgcn_wmma_f32_16x16x4_f32(
                false, a, false, bb, (short)0, acc[t], false, false);
        }
    }
    int rtop = m0 + (lane >> 4) * 8;
#pragma unroll
    for (int t = 0; t < NT; ++t) {
        int c = n0 + t * 16 + (lane & 15);
#pragma unroll
        for (int i = 0; i < 8; ++i)
            Out[(size_t)(rtop + i) * IN_DIM + c] = acc[t][i];
    }
}

// ---------------------------------------------------------------------------
extern "C" void kernel_launch(void* const* d_in, const int* in_sizes, int n_in,
                              void* d_out, int out_size, void* d_ws, size_t ws_size,
                              hipStream_t stream)
{
    const float* x     = (const float*)d_in[0];
    const float* W_in  = (const float*)d_in[1];
    const float* convw = (const float*)d_in[2];
    const float* convb = (const float*)d_in[3];
    const float* W_x   = (const float*)d_in[4];
    const float* W_dt  = (const float*)d_in[5];
    const float* b_dt  = (const float*)d_in[6];
    const float* A_log = (const float*)d_in[7];
    const float* Dv    = (const float*)d_in[8];
    const float* gamma = (const float*)d_in[9];
    const float* beta  = (const float*)d_in[10];
    const float* W_out = (const float*)d_in[11];
    float* out = (float*)d_out;
    float* ws  = (float*)d_ws;

    const size_t SZ = (size_t)NTOK * D_INNER;          // 12.58M floats
    float* xs    = ws;
    float* res   = ws + SZ;
    float* u     = ws + 2 * SZ;
    float* delta = ws + 3 * SZ;
    float* y     = ws + 4 * SZ;
    float* dtl   = ws + 5 * SZ;
    float* Bb    = dtl + (size_t)NTOK * DT_RANK;
    float* Cb    = Bb  + (size_t)NTOK * N_STATE;
    float* yn    = xs;                                  // xs dead after conv
    size_t need = (5 * SZ + (size_t)NTOK * (DT_RANK + 2 * N_STATE)) * sizeof(float);
    if (ws_size < need) return;

    // 1) in-projection (WMMA f32): 2048 m-tiles x 12 n-groups = 24576 waves
    {
        int waves = (NTOK / 16) * (XZ_DIM / 64);
        k_gemm_in<<<waves * 32 / 256, 256, 0, stream>>>(x, W_in, xs, res);
    }
    // 2) conv + SiLU
    {
        long n = (long)SZ;
        k_conv_silu<<<(int)((n + 255) / 256), 256, 0, stream>>>(xs, convw, convb, u);
    }
    // 3) x-projection (N=44)
    {
        long n = (long)NTOK * (DT_RANK + 2 * N_STATE);
        k_xproj<<<(int)((n + 255) / 256), 256, 0, stream>>>(u, W_x, dtl, Bb, Cb);
    }
    // 4) delta = softplus(...)
    {
        long n = (long)SZ;
        k_dt<<<(int)((n + 255) / 256), 256, 0, stream>>>(dtl, W_dt, b_dt, delta);
    }
    // 5) selective scan: 8*384*16 = 49152 lanes
    {
        int n = BATCH * D_INNER * N_STATE;
        k_scan<<<n / 256, 256, 0, stream>>>(delta, u, Bb, Cb, A_log, Dv, y);
    }
    // 6) LayerNorm + gate: wave per token
    {
        k_ln<<<NTOK / 8, 256, 0, stream>>>(y, res, gamma, beta, yn);
    }
    // 7) out-projection (WMMA f32): 2048 x 3 = 6144 waves
    {
        int waves = (NTOK / 16) * (IN_DIM / 64);
        k_gemm_out<<<waves * 32 / 256, 256, 0, stream>>>(yn, W_out, out);
    }
}